// OrnsteinUhlenbeck_9655086481913
// MI455X (gfx1250) — compile-verified
//
#include <hip/hip_runtime.h>

typedef __attribute__((ext_vector_type(2))) float v2f;
typedef __attribute__((ext_vector_type(8))) float v8f;

#define OU_THETA 0.5f

// One wave (32 lanes) per (batch, 16-channel tile). Walks S in 16-step chunks;
// each chunk is x_seg = T @ b' via four v_wmma_f32_16x16x4_f32 K-slices.
// T[i][k] = exp(-theta*(t_i - t_k)) for k <= i (telescoped prefix products),
// carry folded into b'_0 += c_0 * carry.  Carry-dependent K-slice 0 is
// accumulated LAST so the serial cross-chunk chain is one WMMA + one shuffle.
// z loads / x stores are non-temporal (read-once / write-once streams).
template <int DC>
__global__ __launch_bounds__(32)
void ou_scan_wmma(const float* __restrict__ t,
                  const float* __restrict__ x0,
                  const float* __restrict__ z,
                  float* __restrict__ x,
                  int S, int Drt, int tilesPerBatch)
{
    const int D = DC ? DC : Drt;

    const int lane  = threadIdx.x & 31;
    const int wg    = blockIdx.x;
    const int batch = wg / tilesPerBatch;
    const int ctile = wg - batch * tilesPerBatch;
    const int chBase = ctile * 16;
    const int n    = lane & 15;   // channel-in-tile == A-matrix row (M)
    const int half = lane >> 4;   // selects K/N sub-rows per ISA operand layout
    const int ch   = chBase + n;

    const float* tB = t + (size_t)batch * S;
    // per-lane streaming pointers; advanced by one chunk (16*D floats) per iter
    const float* zp = z + (size_t)batch * S * D + ch + (size_t)(2 * half) * D;
    float*       xp = x + (size_t)batch * S * D + ch + (size_t)(8 * half) * D;

    float carry = x0[(size_t)batch * D + ch];   // initial state x0 per channel

    // loop-invariant lower-triangle masks: col = 4*kk + 2*half (+1)
    bool mx[4], my[4];
    #pragma unroll
    for (int kk = 0; kk < 4; ++kk) {
        const int colb = 4 * kk + 2 * half;
        mx[kk] = (colb     <= n);
        my[kk] = (colb + 1 <= n);
    }

    const int nChunks = S >> 4;
    int ti = lane - 1;                 // global t index fetched by this lane
    for (int c = 0; c < nChunks; ++c) {
        // lanes 0..16 provide t[s0-1 .. s0+15]; t_{-1} := 0 (reference prepend)
        int tcl  = ti < 0 ? 0 : (ti >= S ? S - 1 : ti);
        float tt = (ti >= 0) ? tB[tcl] : 0.0f;

        // speculative prefetch of next chunk's z rows
        if (c + 1 < nChunks) {
            __builtin_prefetch(&zp[(size_t)(16 + n - 2 * half) * D], 0, 3);
        }

        const float t_row = __shfl(tt, n + 1, 32);   // t at this lane's row

        v2f Af[4], Bf[4];
        float ca0 = 0.0f;                            // c_0 for the carry fold
        #pragma unroll
        for (int kk = 0; kk < 4; ++kk) {
            const int colb = 4 * kk + 2 * half;           // global step of .x slot
            const float tcm1 = __shfl(tt, colb + 0, 32);  // t[colb-1]
            const float tc0  = __shfl(tt, colb + 1, 32);  // t[colb]
            const float tc1  = __shfl(tt, colb + 2, 32);  // t[colb+1]

            // per-step decay factors c = exp(-theta*dt)
            const float ca = __expf(-OU_THETA * (tc0 - tcm1));
            const float cb = __expf(-OU_THETA * (tc1 - tc0));
            if (kk == 0) ca0 = ca;

            // A slice: one exp, second column derived by telescoping (e1*cb)
            const float e1 = __expf(-OU_THETA * (t_row - tc1));
            Af[kk].x = mx[kk] ? e1 * cb : 0.0f;
            Af[kk].y = my[kk] ? e1      : 0.0f;

            // B slice: b'[step][n] = sqrt(1-c^2) * z  (fast v_sqrt, args in [0,1])
            const float sa = __builtin_amdgcn_sqrtf(fmaxf(0.0f, __builtin_fmaf(-ca, ca, 1.0f)));
            const float sb = __builtin_amdgcn_sqrtf(fmaxf(0.0f, __builtin_fmaf(-cb, cb, 1.0f)));
            Bf[kk].x = sa * __builtin_nontemporal_load(&zp[(size_t)(4 * kk) * D]);
            Bf[kk].y = sb * __builtin_nontemporal_load(&zp[(size_t)(4 * kk + 1) * D]);
        }

        // carry-independent K-slices first
        v8f acc = {};
        acc = __builtin_amdgcn_wmma_f32_16x16x4_f32(false, Af[1], false, Bf[1],
                                                    (short)0, acc, false, false);
        acc = __builtin_amdgcn_wmma_f32_16x16x4_f32(false, Af[2], false, Bf[2],
                                                    (short)0, acc, false, false);
        acc = __builtin_amdgcn_wmma_f32_16x16x4_f32(false, Af[3], false, Bf[3],
                                                    (short)0, acc, false, false);

        // fold carry into step 0 (half==0 lanes' .x slot), then the last slice
        Bf[0].x += (half == 0) ? ca0 * carry : 0.0f;
        acc = __builtin_amdgcn_wmma_f32_16x16x4_f32(false, Af[0], false, Bf[0],
                                                    (short)0, acc, false, false);

        // D layout: VGPR v holds row (v + 8*half), col n -> constant offsets v*D
        #pragma unroll
        for (int v = 0; v < 8; ++v) {
            __builtin_nontemporal_store(acc[v], &xp[(size_t)v * D]);
        }

        // carry = x[15][ch]: row 15 lives in acc[7] of lanes 16..31
        carry = __shfl(acc[7], 16 + n, 32);

        zp += (size_t)16 * D;
        xp += (size_t)16 * D;
        ti += 16;
    }
}

extern "C" void kernel_launch(void* const* d_in, const int* in_sizes, int n_in,
                              void* d_out, int out_size, void* d_ws, size_t ws_size,
                              hipStream_t stream) {
    (void)n_in; (void)out_size; (void)d_ws; (void)ws_size;
    const float* t  = (const float*)d_in[0];  // [B,S,1]
    const float* x0 = (const float*)d_in[1];  // [B,1,D]
    const float* z  = (const float*)d_in[2];  // [B,S,D]
    float* out = (float*)d_out;               // [B,S,D]

    const long long nT  = in_sizes[0];   // B*S
    const long long nX0 = in_sizes[1];   // B*D
    const long long nZ  = in_sizes[2];   // B*S*D

    const int D = (int)(nZ / nT);
    const int B = (int)(nX0 / D);
    const int S = (int)(nT / B);
    const int tilesPerBatch = D / 16;
    const int grid = B * tilesPerBatch;  // 1024 waves for B=64, D=256

    if (D == 256) {
        ou_scan_wmma<256><<<grid, 32, 0, stream>>>(t, x0, z, out, S, D, tilesPerBatch);
    } else {
        ou_scan_wmma<0><<<grid, 32, 0, stream>>>(t, x0, z, out, S, D, tilesPerBatch);
    }
}